// Wav2vec2Loss_69552700391458
// MI455X (gfx1250) — compile-verified
//
#include <hip/hip_runtime.h>
#include <hip/hip_bf16.h>
#include <stdint.h>

#define B_  8
#define T_  1024
#define D_  256
#define K_  100
#define BT_ (B_ * T_)
#define NSTRIPE_ (BT_ / 16)            // 512 16-row stripes

typedef __attribute__((ext_vector_type(2))) float v2f;
typedef __attribute__((ext_vector_type(8))) float v8f;

// ---------------------------------------------------------------------------
// Threefry-2x32 (JAX default PRNG), key = jax.random.key(42) -> (0, 42)
// ---------------------------------------------------------------------------
__device__ __forceinline__ unsigned rotl32(unsigned x, unsigned d) {
    return (x << d) | (x >> (32u - d));
}

__device__ __forceinline__ void tf_round4(unsigned& x0, unsigned& x1,
                                          int r0, int r1, int r2, int r3) {
    x0 += x1; x1 = rotl32(x1, r0); x1 ^= x0;
    x0 += x1; x1 = rotl32(x1, r1); x1 ^= x0;
    x0 += x1; x1 = rotl32(x1, r2); x1 ^= x0;
    x0 += x1; x1 = rotl32(x1, r3); x1 ^= x0;
}

__device__ __forceinline__ uint2 threefry2x32(unsigned k0, unsigned k1,
                                              unsigned x0, unsigned x1) {
    unsigned k2 = k0 ^ k1 ^ 0x1BD11BDAu;
    x0 += k0; x1 += k1;
    tf_round4(x0, x1, 13, 15, 26, 6);  x0 += k1; x1 += k2 + 1u;
    tf_round4(x0, x1, 17, 29, 16, 24); x0 += k2; x1 += k0 + 2u;
    tf_round4(x0, x1, 13, 15, 26, 6);  x0 += k0; x1 += k1 + 3u;
    tf_round4(x0, x1, 17, 29, 16, 24); x0 += k1; x1 += k2 + 4u;
    tf_round4(x0, x1, 13, 15, 26, 6);  x0 += k2; x1 += k0 + 5u;
    return make_uint2(x0, x1);
}

// ---------------------------------------------------------------------------
// Kernel A: per-row inverse norms (inv_nc has 1/TEMP folded in) + zero the
// transposed mask array (grid exactly covers its 512*1024 words; stream order
// guarantees completion before kernel B's atomics).
// ---------------------------------------------------------------------------
__global__ void __launch_bounds__(256) norms_kernel(
    const float* __restrict__ ctx, const float* __restrict__ qf,
    float* __restrict__ inv_nc, float* __restrict__ inv_nq,
    unsigned* __restrict__ maskT) {
    maskT[(size_t)blockIdx.x * 256 + threadIdx.x] = 0u;   // 2048*256 = 512*1024

    const int wave = threadIdx.x >> 5;
    const int lane = threadIdx.x & 31;
    const int row  = blockIdx.x * 8 + wave;               // 0 .. 2*BT_-1
    const float* src = (row < BT_) ? (ctx + (size_t)row * D_)
                                   : (qf + (size_t)(row - BT_) * D_);
    float s = 0.f;
    for (int c = lane; c < D_; c += 32) { float v = src[c]; s += v * v; }
    #pragma unroll
    for (int off = 16; off > 0; off >>= 1) s += __shfl_xor(s, off, 32);
    if (lane == 0) {
        float inv = 1.0f / fmaxf(sqrtf(s), 1e-8f);
        if (row < BT_) inv_nc[row] = inv * 10.0f;         // fold 1/TEMP
        else           inv_nq[row - BT_] = inv;
    }
}

// ---------------------------------------------------------------------------
// Kernel B: exact _sample_neg_indices via threefry + stable bitonic selection.
// Emits a column-major membership mask: maskT[stripe][col] bit (t&15) set iff
// column col is in row t's 101-sample set (100 negatives + diagonal positive).
// ---------------------------------------------------------------------------
__global__ void __launch_bounds__(256) negmask_kernel(unsigned* __restrict__ maskT) {
    __shared__ unsigned long long keys[1024];
    const int n   = blockIdx.x;                // b*T + t
    const int t   = n & (T_ - 1);
    const int tid = threadIdx.x;
    const unsigned Ntot = (unsigned)BT_ * (unsigned)(T_ - 1);   // 8,380,416
    const unsigned Nh   = Ntot >> 1;                            // split-in-half rule

    #pragma unroll
    for (int s = 0; s < 4; ++s) {
        int j = tid + s * 256;
        unsigned long long key;
        if (j < T_ - 1) {
            unsigned p = (unsigned)n * (unsigned)(T_ - 1) + (unsigned)j;
            unsigned bits;
            if (p < Nh) { uint2 r = threefry2x32(0u, 42u, p, p + Nh); bits = r.x; }
            else        { uint2 r = threefry2x32(0u, 42u, p - Nh, p); bits = r.y; }
            // jax.random.uniform is monotone in (bits>>9); append j for stable ties
            key = ((unsigned long long)(bits >> 9) << 10) | (unsigned long long)j;
        } else {
            key = ~0ull;                       // pad slot sorts to the end
        }
        keys[j] = key;
    }
    __syncthreads();

    for (int k = 2; k <= 1024; k <<= 1) {      // bitonic sort, ascending
        for (int jj = k >> 1; jj > 0; jj >>= 1) {
            #pragma unroll
            for (int s = 0; s < 4; ++s) {
                int i = tid + s * 256;
                int ixj = i ^ jj;
                if (ixj > i) {
                    bool up = ((i & k) == 0);
                    unsigned long long a = keys[i], b = keys[ixj];
                    if ((a > b) == up) { keys[i] = b; keys[ixj] = a; }
                }
            }
            __syncthreads();
        }
    }

    const size_t stripeBase = (size_t)(n >> 4) * T_;   // stripe = b*64 + t/16
    const unsigned rowbit   = 1u << (t & 15);
    if (tid < K_) {
        int j  = (int)(keys[tid] & 1023ull);
        int jj = j + (j >= t ? 1 : 0);         // skip-self adjustment
        atomicOr(&maskT[stripeBase + jj], rowbit);
    }
    if (tid == K_) {
        atomicOr(&maskT[stripeBase + t], rowbit);       // positive (diagonal) bit
    }
}

// ---------------------------------------------------------------------------
// Kernel C: fused WMMA GEMM (Sim = C·Qᵀ per batch) + masked online logsumexp.
// One block per 16-row stripe; 8 wave32s; each wave runs TWO independent
// 16x16 accumulator chains (column tiles jt and jt+32) sharing the A fragment.
// Epilogue is straight-line VALU: one LDS mask-word load per tile, cndmask
// diagonal capture, branchless online (max,sumexp) update.
// ---------------------------------------------------------------------------
__global__ void __launch_bounds__(256) sims_lse_kernel(
    const float* __restrict__ ctx, const float* __restrict__ qf,
    const float* __restrict__ inv_nc, const float* __restrict__ inv_nq,
    const unsigned* __restrict__ maskT, float* __restrict__ partials) {
    __shared__ float    ctx_s[16][258];     // +2 pad to spread LDS banks
    __shared__ float    invq_s[T_];
    __shared__ unsigned maskc_s[T_];        // 16 row-bits per column
    __shared__ float    invc_s[16];
    __shared__ float    pos_s[16];
    __shared__ float    part_mx[8][16];
    __shared__ float    part_sm[8][16];
    __shared__ float    red_s[16];

    const int tid  = threadIdx.x;
    const int wave = tid >> 5, lane = tid & 31;
    const int kh   = lane >> 4, ncol = lane & 15;
    const int b    = blockIdx.x >> 6;
    const int t0   = (blockIdx.x & 63) << 4;
    const size_t cbase = ((size_t)b * T_ + t0) * D_;
    const size_t qbase = (size_t)b * T_ * D_;

    for (int idx = tid; idx < 16 * D_; idx += 256) {
        int r = idx >> 8, c = idx & 255;
        ctx_s[r][c] = ctx[cbase + (size_t)r * D_ + c];
    }
    for (int i = tid; i < T_; i += 256) {
        invq_s[i]  = inv_nq[(size_t)b * T_ + i];
        maskc_s[i] = maskT[(size_t)blockIdx.x * T_ + i];
    }
    if (tid < 16) { invc_s[tid] = inv_nc[(size_t)b * T_ + t0 + tid]; pos_s[tid] = 0.f; }
    __syncthreads();

    float invc_r[8];
    #pragma unroll
    for (int v = 0; v < 8; ++v) invc_r[v] = invc_s[v + 8 * kh];

    float mx[8], sm[8];
    #pragma unroll
    for (int v = 0; v < 8; ++v) { mx[v] = -1e30f; sm[v] = 0.f; }

    const int  vsel   = ncol - 8 * kh;               // which v holds this lane's diag
    const bool mydiag = (wave == ((t0 >> 4) & 7));   // this wave owns the diag tile
    float posr = 0.f;

    // branchless masked online-logsumexp epilogue for one 16x16 tile
    auto epilogue = [&](int j0, const v8f& acc, bool isdiag) {
        const int      col    = j0 + ncol;
        const float    invq_c = invq_s[col];
        const unsigned mword  = maskc_s[col] >> (8 * kh);
        #pragma unroll
        for (int v = 0; v < 8; ++v) {
            float val = acc[v] * invc_r[v] * invq_c;           // cosine / TEMP
            posr = (isdiag && v == vsel) ? val : posr;         // diag capture (cndmask)
            unsigned bit = (mword >> v) & 1u;
            float vv = bit ? val : -1e30f;
            float nm = fmaxf(mx[v], vv);
            sm[v] = sm[v] * __expf(mx[v] - nm) + (float)bit * __expf(vv - nm);
            mx[v] = nm;
        }
    };

    for (int jt = wave; jt < 32; jt += 8) {    // wave-uniform: EXEC all-ones at WMMA
        const int j0a = jt << 4;
        const int j0b = (jt + 32) << 4;
        v8f acc0 = {0.f, 0.f, 0.f, 0.f, 0.f, 0.f, 0.f, 0.f};
        v8f acc1 = {0.f, 0.f, 0.f, 0.f, 0.f, 0.f, 0.f, 0.f};
        const float* qrow0 = qf + qbase + (size_t)(j0a + ncol) * D_ + 2 * kh;
        const float* qrow1 = qf + qbase + (size_t)(j0b + ncol) * D_ + 2 * kh;
        #pragma unroll 4
        for (int k0 = 0; k0 < D_; k0 += 4) {
            v2f a  = *(const v2f*)&ctx_s[ncol][k0 + 2 * kh];   // A: row m=l&15 (shared)
            v2f b0 = *(const v2f*)&qrow0[k0];                  // B: col n=l&15
            v2f b1 = *(const v2f*)&qrow1[k0];
            acc0 = __builtin_amdgcn_wmma_f32_16x16x4_f32(
                false, a, false, b0, (short)0, acc0, false, false);
            acc1 = __builtin_amdgcn_wmma_f32_16x16x4_f32(
                false, a, false, b1, (short)0, acc1, false, false);
        }
        epilogue(j0a, acc0, j0a == t0);
        epilogue(j0b, acc1, j0b == t0);
    }

    if (mydiag && vsel >= 0 && vsel < 8) pos_s[ncol] = posr;   // one store per diag lane

    // merge (max,sumexp) across the 16 lanes of each half-wave (same rows)
    #pragma unroll
    for (int v = 0; v < 8; ++v) {
        #pragma unroll
        for (int off = 1; off < 16; off <<= 1) {
            float omx = __shfl_xor(mx[v], off, 32);
            float osm = __shfl_xor(sm[v], off, 32);
            float nm  = fmaxf(mx[v], omx);
            sm[v] = sm[v] * __expf(mx[v] - nm) + osm * __expf(omx - nm);
            mx[v] = nm;
        }
    }
    if (ncol == 0) {                           // lanes 0 (rows 0-7), 16 (rows 8-15)
        #pragma unroll
        for (int v = 0; v < 8; ++v) {
            part_mx[wave][v + 8 * kh] = mx[v];
            part_sm[wave][v + 8 * kh] = sm[v];
        }
    }
    __syncthreads();

    if (tid < 16) {                            // merge the 8 wave partials per row
        float M = -1e30f, S = 0.f;
        for (int w = 0; w < 8; ++w) {
            float omx = part_mx[w][tid], osm = part_sm[w][tid];
            float nm  = fmaxf(M, omx);
            S = S * __expf(M - nm) + osm * __expf(omx - nm);
            M = nm;
        }
        red_s[tid] = (__logf(S) + M) - pos_s[tid];    // lse - sims[:,0]
    }
    __syncthreads();
    if (tid == 0) {
        float s = 0.f;
        #pragma unroll
        for (int i = 0; i < 16; ++i) s += red_s[i];
        partials[blockIdx.x] = s;              // 512 deterministic partials
    }
}

// ---------------------------------------------------------------------------
// Finalize: fixed-order tree reduction of the 512 partials (deterministic).
// ---------------------------------------------------------------------------
__global__ void __launch_bounds__(256) finalize_kernel(
    const float* __restrict__ partials, const float* __restrict__ divloss,
    float* __restrict__ out) {
    __shared__ float buf[256];
    const int tid = threadIdx.x;
    buf[tid] = partials[tid] + partials[tid + 256];
    __syncthreads();
    for (int off = 128; off > 0; off >>= 1) {
        if (tid < off) buf[tid] += buf[tid + off];
        __syncthreads();
    }
    if (tid == 0) out[0] = buf[0] / (float)BT_ + 0.4f * divloss[0];
}

// ---------------------------------------------------------------------------
extern "C" void kernel_launch(void* const* d_in, const int* in_sizes, int n_in,
                              void* d_out, int out_size, void* d_ws, size_t ws_size,
                              hipStream_t stream) {
    const float* ctx = (const float*)d_in[0];   // context_repr (B,T,D) f32
    const float* qf  = (const float*)d_in[1];   // quantized_features (B,T,D) f32
    const float* dl  = (const float*)d_in[2];   // diversity_loss scalar
    (void)in_sizes; (void)n_in; (void)out_size; (void)ws_size;
    // d_in[3] (time_mask) is unused by the reference computation.

    float*    inv_nc   = (float*)d_ws;                          // 8192 f32
    float*    inv_nq   = inv_nc + BT_;                          // 8192 f32
    unsigned* maskT    = (unsigned*)(inv_nq + BT_);             // 512*1024 u32 (2 MB)
    float*    partials = (float*)(maskT + (size_t)NSTRIPE_ * T_); // 512 f32

    norms_kernel   <<<2048, 256, 0, stream>>>(ctx, qf, inv_nc, inv_nq, maskT);
    negmask_kernel <<<BT_,  256, 0, stream>>>(maskT);
    sims_lse_kernel<<<512,  256, 0, stream>>>(ctx, qf, inv_nc, inv_nq, maskT, partials);
    finalize_kernel<<<1,    256, 0, stream>>>(partials, dl, (float*)d_out);
}